// GINConv_53163105190017
// MI455X (gfx1250) — compile-verified
//
#include <hip/hip_runtime.h>

#define N_NODES 50000
#define N_EDGES 800000
#define D       64            // D_IN == D_OUT == 64

typedef __attribute__((ext_vector_type(2))) float v2f;
typedef __attribute__((ext_vector_type(8))) float v8f;

// ---------------------------------------------------------------------------
// Kernel 1: rst[i] = (1 + eps) * feat[i]          (float4 vectorized)
// ---------------------------------------------------------------------------
__global__ void gin_init_kernel(const float* __restrict__ feat,
                                const float* __restrict__ eps,
                                float* __restrict__ rst) {
    const int total4 = N_NODES * D / 4;
    int i = blockIdx.x * blockDim.x + threadIdx.x;
    if (i >= total4) return;
    const float s = 1.0f + eps[0];
    float4 v = ((const float4*)feat)[i];
    v.x *= s; v.y *= s; v.z *= s; v.w *= s;
    ((float4*)rst)[i] = v;
}

// ---------------------------------------------------------------------------
// Kernel 2: for each edge e: rst[dst[e]] += feat[src[e]]   (hw f32 atomics)
// One thread per (edge, 4-float group). 16 consecutive lanes share one edge,
// so the float4 gathers of feat[src] rows are fully coalesced (256B/row) and
// feat (12.8 MB) stays resident in the 192 MB L2.
// ---------------------------------------------------------------------------
__global__ void gin_scatter_kernel(const float* __restrict__ feat,
                                   const int* __restrict__ src,
                                   const int* __restrict__ dst,
                                   float* __restrict__ rst) {
    long long t = (long long)blockIdx.x * blockDim.x + threadIdx.x;
    if (t >= (long long)N_EDGES * (D / 4)) return;
    int e = (int)(t >> 4);      // edge index
    int g = (int)(t & 15);      // which float4 of the 64-float row
    int s = src[e];
    int d = dst[e];
    float4 v = ((const float4*)(feat + (size_t)s * D))[g];
    float* p = rst + (size_t)d * D + (size_t)g * 4;
    unsafeAtomicAdd(p + 0, v.x);   // -> global_atomic_add_f32
    unsafeAtomicAdd(p + 1, v.y);
    unsafeAtomicAdd(p + 2, v.z);
    unsafeAtomicAdd(p + 3, v.w);
}

// ---------------------------------------------------------------------------
// Kernel 3: out = rst @ W^T + b via V_WMMA_F32_16X16X4_F32 (full fp32).
//   grid.x  = N_NODES/16 = 3125 (M-tiles of 16 rows, exact)
//   block   = 128 threads = 4 wave32; wave w computes N-tile w (cols w*16..)
//   K = 64 -> 16 chained WMMA ops, accumulator in v8f.
//
// Fragment layouts (CDNA5 ISA, 16x16x4 f32):
//   A (16x4):  lane l: M = l&15, holds K = kb + (l>>4)*2 + {0,1} in 2 VGPRs
//   B (4x16):  lane l: N = l&15, holds K = kb + (l>>4)*2 + {0,1}
//              B[k][n] = W[n][k]  (out = rst @ W.T, torch Linear layout)
//   C/D:       vgpr r, lane l -> element (M = r + (l>>4)*8, N = l&15)
// ---------------------------------------------------------------------------
__global__ void gin_gemm_wmma_kernel(const float* __restrict__ rst,
                                     const float* __restrict__ W,
                                     const float* __restrict__ bias,
                                     float* __restrict__ out) {
    const int lane  = threadIdx.x & 31;
    const int wave  = threadIdx.x >> 5;      // 0..3 -> N tile
    const int mtile = blockIdx.x;

    const int m    = lane & 15;
    const int koff = (lane >> 4) * 2;        // upper half-wave holds K+2,K+3
    const int ng   = wave * 16 + (lane & 15);

    const float* __restrict__ Arow = rst + ((size_t)mtile * 16 + m) * D;
    const float* __restrict__ Brow = W + (size_t)ng * D;   // row n of W == col n of B

    v8f c = {0.f, 0.f, 0.f, 0.f, 0.f, 0.f, 0.f, 0.f};

#pragma unroll
    for (int kb = 0; kb < D; kb += 4) {
        v2f a, b;
        a.x = Arow[kb + koff];
        a.y = Arow[kb + koff + 1];
        b.x = Brow[kb + koff];
        b.y = Brow[kb + koff + 1];
        // 8 args: (neg_a, A, neg_b, B, c_mod, C, reuse_a, reuse_b)
        c = __builtin_amdgcn_wmma_f32_16x16x4_f32(false, a, false, b,
                                                  (short)0, c, false, false);
    }

    const float bv = bias[ng];
#pragma unroll
    for (int r = 0; r < 8; ++r) {
        int mm = r + (lane >> 4) * 8;
        out[((size_t)mtile * 16 + mm) * D + ng] = c[r] + bv;
    }
}

// ---------------------------------------------------------------------------
// Launcher
// inputs: feat[50000*64] f32, W[64*64] f32, b[64] f32, eps[1] f32,
//         src[800000] i32, dst[800000] i32
// output: out[50000*64] f32
// d_ws:   rst accumulator, 50000*64*4 = 12.8 MB
// ---------------------------------------------------------------------------
extern "C" void kernel_launch(void* const* d_in, const int* in_sizes, int n_in,
                              void* d_out, int out_size, void* d_ws, size_t ws_size,
                              hipStream_t stream) {
    (void)in_sizes; (void)n_in; (void)out_size; (void)ws_size;

    const float* feat = (const float*)d_in[0];
    const float* W    = (const float*)d_in[1];
    const float* bias = (const float*)d_in[2];
    const float* eps  = (const float*)d_in[3];
    const int*   src  = (const int*)d_in[4];
    const int*   dst  = (const int*)d_in[5];
    float*       out  = (float*)d_out;
    float*       rst  = (float*)d_ws;

    // 1) rst = (1+eps)*feat   (800000 float4 elements)
    {
        int total4 = N_NODES * D / 4;
        int blocks = (total4 + 255) / 256;
        gin_init_kernel<<<blocks, 256, 0, stream>>>(feat, eps, rst);
    }
    // 2) scatter-add edges (12.8M threads, 4 hw atomics each)
    {
        long long total = (long long)N_EDGES * (D / 4);
        int blocks = (int)((total + 255) / 256);
        gin_scatter_kernel<<<blocks, 256, 0, stream>>>(feat, src, dst, rst);
    }
    // 3) WMMA GEMM + bias (3125 blocks x 128 threads)
    {
        gin_gemm_wmma_kernel<<<N_NODES / 16, 128, 0, stream>>>(rst, W, bias, out);
    }
}